// KGCLNDA_28475633173135
// MI455X (gfx1250) — compile-verified
//
#include <hip/hip_runtime.h>

// ---------------------------------------------------------------------------
// CDNA5 (gfx1250) implementation. wave32. All GEMMs via V_WMMA_F32_16X16X4_F32.
// ---------------------------------------------------------------------------

typedef __attribute__((ext_vector_type(2))) float v2f;
typedef __attribute__((ext_vector_type(8))) float v8f;

__device__ __forceinline__ v8f wmma_f32(v2f a, v2f b, v8f c) {
  // 8 args: (neg_a, A, neg_b, B, c_mod, C, reuse_a, reuse_b)
  return __builtin_amdgcn_wmma_f32_16x16x4_f32(false, a, false, b, (short)0, c,
                                               false, false);
}

// Problem constants (match reference setup_inputs)
static constexpr int Nn   = 100000;   // graph nodes
static constexpr int Dd   = 128;      // embedding dim
static constexpr int NNZ  = 2000000;
static constexpr int Bb   = 32;
static constexpr int Ll   = 256;
static constexpr int NLl  = 2;
static constexpr int DIi  = 512;
static constexpr int NH   = 8;
static constexpr int DK   = 16;
static constexpr int TOK  = Bb * Ll;  // 8192 tokens

// ---------------------------------------------------------------------------
// zero fill
// ---------------------------------------------------------------------------
__global__ void zero_kernel(float* __restrict__ p, int n) {
  int i = blockIdx.x * blockDim.x + threadIdx.x;
  if (i < n) p[i] = 0.0f;
}

// ---------------------------------------------------------------------------
// COO SpMM: side[r] += vals[e] * ego[cols[e]]  (rows sorted ascending)
// Block = 128 threads (one per dim). Each block walks CHUNK nnz serially,
// accumulating in registers while the row is unchanged; atomic flush on
// row change / chunk end. Sorted rows => ~avg 20 nnz per row => few atomics.
// ---------------------------------------------------------------------------
static constexpr int SPMM_CHUNK = 32;

__global__ void spmm_kernel(const float* __restrict__ ego,
                            const float* __restrict__ vals,
                            const int* __restrict__ rows,
                            const int* __restrict__ cols,
                            float* __restrict__ side) {
  int start = blockIdx.x * SPMM_CHUNK;
  int d = threadIdx.x;  // 0..127
  float acc = 0.0f;
  int cur = rows[start];
  for (int e = start; e < start + SPMM_CHUNK; ++e) {
    int r = rows[e];                 // uniform across block
    if (r != cur) {                  // uniform branch
      atomicAdd(&side[(size_t)cur * Dd + d], acc);
      acc = 0.0f;
      cur = r;
    }
    acc = fmaf(vals[e], ego[(size_t)cols[e] * Dd + d], acc);
  }
  atomicAdd(&side[(size_t)cur * Dd + d], acc);
}

// ---------------------------------------------------------------------------
// Graph aggregator:
//   agg = leaky((ego+side)@W1 + b1) + leaky((ego*side)@W2 + b2)
// One wave per 16x16 output tile, dual WMMA accumulators.
// ---------------------------------------------------------------------------
__global__ __launch_bounds__(256) void agg_kernel(
    const float* __restrict__ ego, const float* __restrict__ side,
    const float* __restrict__ W1, const float* __restrict__ b1,
    const float* __restrict__ W2, const float* __restrict__ b2,
    float* __restrict__ out) {
  const int ntiles = Dd / 16;                       // 8
  int wave = blockIdx.x * (blockDim.x >> 5) + (threadIdx.x >> 5);
  int total = (Nn / 16) * ntiles;                   // 50000
  if (wave >= total) return;                        // wave-uniform
  int lane = threadIdx.x & 31;
  int lane16 = lane & 15;
  int half = lane >> 4;
  int kb = half * 2;

  int mt = wave / ntiles;
  int nt = wave - mt * ntiles;
  int mrow = mt * 16 + lane16;
  int ncol = nt * 16 + lane16;

  v8f acc1 = {}, acc2 = {};
  const float* er = ego + (size_t)mrow * Dd;
  const float* sr = side + (size_t)mrow * Dd;
  for (int k0 = 0; k0 < Dd; k0 += 4) {
    float e0 = er[k0 + kb], s0 = sr[k0 + kb];
    float e1 = er[k0 + kb + 1], s1 = sr[k0 + kb + 1];
    v2f a1, a2, bw1, bw2;
    a1.x = e0 + s0; a1.y = e1 + s1;                 // sum branch
    a2.x = e0 * s0; a2.y = e1 * s1;                 // bi branch
    bw1.x = W1[(k0 + kb) * Dd + ncol];
    bw1.y = W1[(k0 + kb + 1) * Dd + ncol];
    bw2.x = W2[(k0 + kb) * Dd + ncol];
    bw2.y = W2[(k0 + kb + 1) * Dd + ncol];
    acc1 = wmma_f32(a1, bw1, acc1);
    acc2 = wmma_f32(a2, bw2, acc2);
  }
  float bn1 = b1[ncol], bn2 = b2[ncol];
#pragma unroll
  for (int i = 0; i < 8; ++i) {
    int m = mt * 16 + half * 8 + i;
    float v1 = acc1[i] + bn1;
    float v2 = acc2[i] + bn2;
    v1 = (v1 > 0.0f) ? v1 : 0.01f * v1;             // leaky_relu slope 0.01
    v2 = (v2 > 0.0f) ? v2 : 0.01f * v2;
    out[(size_t)m * Dd + ncol] = v1 + v2;
  }
}

// ---------------------------------------------------------------------------
// Generic GEMM: C = act(A[M,K] @ W[K,N] + bias[N] + (res? res[M,N]:0))
// ACT: 0 = none, 1 = relu. One wave per 16x16 tile.
// ---------------------------------------------------------------------------
template <int ACT>
__global__ __launch_bounds__(256) void gemm_bias_kernel(
    const float* __restrict__ A, const float* __restrict__ W,
    const float* __restrict__ bias, const float* __restrict__ res,
    float* __restrict__ C, int M, int K, int N) {
  int wave = blockIdx.x * (blockDim.x >> 5) + (threadIdx.x >> 5);
  int ntiles = N >> 4;
  int total = (M >> 4) * ntiles;
  if (wave >= total) return;                        // wave-uniform
  int lane = threadIdx.x & 31;
  int lane16 = lane & 15;
  int half = lane >> 4;
  int kb = half * 2;

  int mt = wave / ntiles;
  int nt = wave - mt * ntiles;
  int mrow = mt * 16 + lane16;
  int ncol = nt * 16 + lane16;

  v8f acc = {};
  const float* ar = A + (size_t)mrow * K;
  for (int k0 = 0; k0 < K; k0 += 4) {
    v2f a, b;
    a.x = ar[k0 + kb];
    a.y = ar[k0 + kb + 1];
    b.x = W[(size_t)(k0 + kb) * N + ncol];
    b.y = W[(size_t)(k0 + kb + 1) * N + ncol];
    acc = wmma_f32(a, b, acc);
  }
  float bn = bias[ncol];
#pragma unroll
  for (int i = 0; i < 8; ++i) {
    int m = mt * 16 + half * 8 + i;
    float v = acc[i] + bn;
    if (res) v += res[(size_t)m * N + ncol];
    if (ACT == 1) v = fmaxf(v, 0.0f);
    C[(size_t)m * N + ncol] = v;
  }
}

// ---------------------------------------------------------------------------
// Fused attention, one wave per (b, h, 16-row tile):
//   scores(16x256) via 16 WMMA C-tiles held in registers -> register softmax
//   (shfl 16-lane reductions) -> probs staged to LDS (stride-padded) ->
//   attn@V via WMMA -> o (B,L,D).
// q,k,v layout: (B, L, H, DK) contiguous == (B, L, 128).
// ---------------------------------------------------------------------------
static constexpr int PB_STRIDE = 257;  // 16x256 tile, pad to kill bank conflicts

__global__ __launch_bounds__(64) void attn_kernel(const float* __restrict__ q,
                                                  const float* __restrict__ k,
                                                  const float* __restrict__ v,
                                                  float* __restrict__ o) {
  __shared__ float probs[2][16 * PB_STRIDE];
  int wid = threadIdx.x >> 5;
  int wg = blockIdx.x * 2 + wid;      // 0..4095
  int rt = wg & 15;                   // row tile within sequence
  int h = (wg >> 4) & 7;
  int b = wg >> 7;
  int lane = threadIdx.x & 31;
  int lane16 = lane & 15;
  int half = lane >> 4;
  int kb = half * 2;
  const float scale = 0.25f;          // 1/sqrt(16)

  // ---- scores = (Q Kt) * scale : 16 tiles of 16x16 ----
  v8f sc[16];
#pragma unroll
  for (int t = 0; t < 16; ++t) sc[t] = v8f{};

  size_t qrow = ((size_t)(b * Ll + rt * 16 + lane16)) * Dd + h * DK;
#pragma unroll
  for (int t = 0; t < 16; ++t) {
    size_t krow = ((size_t)(b * Ll + t * 16 + lane16)) * Dd + h * DK;
#pragma unroll
    for (int k0 = 0; k0 < DK; k0 += 4) {
      v2f A, B;
      A.x = q[qrow + k0 + kb];
      A.y = q[qrow + k0 + kb + 1];
      B.x = k[krow + k0 + kb];        // Kt: B[c][n] = K[n][c]
      B.y = k[krow + k0 + kb + 1];
      sc[t] = wmma_f32(A, B, sc[t]);
    }
  }

  // ---- register softmax over 256 cols per row ----
#pragma unroll
  for (int j = 0; j < 8; ++j) {
    float mx = -1e30f;
#pragma unroll
    for (int t = 0; t < 16; ++t) mx = fmaxf(mx, sc[t][j] * scale);
#pragma unroll
    for (int s = 1; s < 16; s <<= 1) mx = fmaxf(mx, __shfl_xor(mx, s));
    float sum = 0.0f;
#pragma unroll
    for (int t = 0; t < 16; ++t) {
      float e = __expf(sc[t][j] * scale - mx);
      sc[t][j] = e;
      sum += e;
    }
#pragma unroll
    for (int s = 1; s < 16; s <<= 1) sum += __shfl_xor(sum, s);
    float inv = 1.0f / sum;
#pragma unroll
    for (int t = 0; t < 16; ++t) sc[t][j] *= inv;
  }

  // ---- stage probs to LDS (C-layout -> row-major) ----
  float* pb = probs[wid];
#pragma unroll
  for (int t = 0; t < 16; ++t) {
#pragma unroll
    for (int j = 0; j < 8; ++j) {
      pb[(half * 8 + j) * PB_STRIDE + t * 16 + lane16] = sc[t][j];
    }
  }

  // ---- o_tile(16x16) = probs(16x256) @ V(256x16) ----
  v8f oa = {};
  for (int k0 = 0; k0 < Ll; k0 += 4) {
    v2f A, B;
    A.x = pb[lane16 * PB_STRIDE + k0 + kb];
    A.y = pb[lane16 * PB_STRIDE + k0 + kb + 1];
    B.x = v[((size_t)(b * Ll + k0 + kb)) * Dd + h * DK + lane16];
    B.y = v[((size_t)(b * Ll + k0 + kb + 1)) * Dd + h * DK + lane16];
    oa = wmma_f32(A, B, oa);
  }
#pragma unroll
  for (int i = 0; i < 8; ++i) {
    int m = rt * 16 + half * 8 + i;
    o[((size_t)(b * Ll + m)) * Dd + h * DK + lane16] = oa[i];
  }
}

// ---------------------------------------------------------------------------
// LayerNorm over D=128, one wave per row (4 elems/lane + shfl reduction).
// ---------------------------------------------------------------------------
__global__ __launch_bounds__(256) void ln_kernel(const float* __restrict__ in,
                                                 const float* __restrict__ g,
                                                 const float* __restrict__ be,
                                                 float* __restrict__ out,
                                                 int rows) {
  int wave = blockIdx.x * (blockDim.x >> 5) + (threadIdx.x >> 5);
  if (wave >= rows) return;
  int lane = threadIdx.x & 31;
  const float* r = in + (size_t)wave * Dd;
  float x0 = r[lane], x1 = r[lane + 32], x2 = r[lane + 64], x3 = r[lane + 96];
  float s = x0 + x1 + x2 + x3;
#pragma unroll
  for (int m = 1; m < 32; m <<= 1) s += __shfl_xor(s, m);
  float mean = s * (1.0f / 128.0f);
  float d0 = x0 - mean, d1 = x1 - mean, d2 = x2 - mean, d3 = x3 - mean;
  float ss = d0 * d0 + d1 * d1 + d2 * d2 + d3 * d3;
#pragma unroll
  for (int m = 1; m < 32; m <<= 1) ss += __shfl_xor(ss, m);
  float inv = rsqrtf(ss * (1.0f / 128.0f) + 1e-5f);
  float* w = out + (size_t)wave * Dd;
  w[lane]      = d0 * inv * g[lane]      + be[lane];
  w[lane + 32] = d1 * inv * g[lane + 32] + be[lane + 32];
  w[lane + 64] = d2 * inv * g[lane + 64] + be[lane + 64];
  w[lane + 96] = d3 * inv * g[lane + 96] + be[lane + 96];
}

// ---------------------------------------------------------------------------
// Host orchestration
// ---------------------------------------------------------------------------
extern "C" void kernel_launch(void* const* d_in, const int* in_sizes, int n_in,
                              void* d_out, int out_size, void* d_ws,
                              size_t ws_size, hipStream_t stream) {
  (void)in_sizes; (void)n_in; (void)out_size; (void)ws_size;
  const float* ego   = (const float*)d_in[0];
  const float* vals  = (const float*)d_in[1];
  const float* W1    = (const float*)d_in[2];
  const float* b1    = (const float*)d_in[3];
  const float* W2    = (const float*)d_in[4];
  const float* b2    = (const float*)d_in[5];
  const float* encin = (const float*)d_in[6];
  const float* wq    = (const float*)d_in[7];
  const float* bq    = (const float*)d_in[8];
  const float* wk    = (const float*)d_in[9];
  const float* bk    = (const float*)d_in[10];
  const float* wv    = (const float*)d_in[11];
  const float* bv    = (const float*)d_in[12];
  const float* wo    = (const float*)d_in[13];
  const float* bo    = (const float*)d_in[14];
  const float* ln1g  = (const float*)d_in[15];
  const float* ln1b  = (const float*)d_in[16];
  const float* cw1   = (const float*)d_in[17];
  const float* cb1   = (const float*)d_in[18];
  const float* cw2   = (const float*)d_in[19];
  const float* cb2   = (const float*)d_in[20];
  const float* ln2g  = (const float*)d_in[21];
  const float* ln2b  = (const float*)d_in[22];
  const int*   rows  = (const int*)d_in[23];
  const int*   cols  = (const int*)d_in[24];
  float* out = (float*)d_out;

  const size_t ND  = (size_t)Nn * Dd;       // 12.8M floats
  const size_t BLD = (size_t)TOK * Dd;      // 1.05M floats
  float* ws = (float*)d_ws;
  float* side = ws;                   // ND
  float* xbuf = side + ND;            // BLD
  float* qb   = xbuf + BLD;           // BLD
  float* kb_  = qb + BLD;             // BLD
  float* vb   = kb_ + BLD;            // BLD
  float* ab   = vb + BLD;             // BLD (attention output)
  float* tb   = ab + BLD;             // BLD (pre-LN temp)
  float* hb   = tb + BLD;             // TOK*DI (FFN hidden)

  // ---------------- graph path ----------------
  zero_kernel<<<(int)((ND + 255) / 256), 256, 0, stream>>>(side, (int)ND);
  spmm_kernel<<<NNZ / SPMM_CHUNK, Dd, 0, stream>>>(ego, vals, rows, cols, side);
  {
    int tiles = (Nn / 16) * (Dd / 16);          // 50000 waves
    agg_kernel<<<(tiles + 7) / 8, 256, 0, stream>>>(ego, side, W1, b1, W2, b2,
                                                    out);
  }

  // ---------------- transformer encoder ----------------
  auto gemm_blocks = [](int M, int N) { return ((M / 16) * (N / 16) + 7) / 8; };
  for (int i = 0; i < NLl; ++i) {
    const float* xin = (i == 0) ? encin : xbuf;
    const size_t wOff = (size_t)i * Dd * Dd;
    const size_t bOff = (size_t)i * Dd;
    // QKV projections
    gemm_bias_kernel<0><<<gemm_blocks(TOK, Dd), 256, 0, stream>>>(
        xin, wq + wOff, bq + bOff, nullptr, qb, TOK, Dd, Dd);
    gemm_bias_kernel<0><<<gemm_blocks(TOK, Dd), 256, 0, stream>>>(
        xin, wk + wOff, bk + bOff, nullptr, kb_, TOK, Dd, Dd);
    gemm_bias_kernel<0><<<gemm_blocks(TOK, Dd), 256, 0, stream>>>(
        xin, wv + wOff, bv + bOff, nullptr, vb, TOK, Dd, Dd);
    // fused attention: 4096 waves, 2 per block
    attn_kernel<<<Bb * NH * (Ll / 16) / 2, 64, 0, stream>>>(qb, kb_, vb, ab);
    // output projection + residual, then LN1
    gemm_bias_kernel<0><<<gemm_blocks(TOK, Dd), 256, 0, stream>>>(
        ab, wo + wOff, bo + bOff, xin, tb, TOK, Dd, Dd);
    ln_kernel<<<TOK / 8, 256, 0, stream>>>(tb, ln1g + bOff, ln1b + bOff, xbuf,
                                           TOK);
    // FFN: relu(x@cw1+cb1) @ cw2 + cb2 + x, then LN2
    gemm_bias_kernel<1><<<gemm_blocks(TOK, DIi), 256, 0, stream>>>(
        xbuf, cw1 + (size_t)i * Dd * DIi, cb1 + (size_t)i * DIi, nullptr, hb,
        TOK, Dd, DIi);
    gemm_bias_kernel<0><<<gemm_blocks(TOK, Dd), 256, 0, stream>>>(
        hb, cw2 + (size_t)i * DIi * Dd, cb2 + bOff, xbuf, tb, TOK, DIi, Dd);
    float* lnout = (i == NLl - 1) ? (out + ND) : xbuf;
    ln_kernel<<<TOK / 8, 256, 0, stream>>>(tb, ln2g + bOff, ln2b + bOff, lnout,
                                           TOK);
  }
}